// RankInfoNCELoss_11218454577734
// MI455X (gfx1250) — compile-verified
//
#include <hip/hip_runtime.h>
#include <hip/hip_bf16.h>
#include <math.h>

typedef __attribute__((ext_vector_type(2))) float v2f;
typedef __attribute__((ext_vector_type(8))) float v8f;

#define B_N 4096
#define D_K 1024
#define NQ_ 1024
#define INV_TEMP 10.0f
#define RANK_W 0.1f
#define SCORE_W 0.01f
#define NCHUNK 16           // 16 column chunks of 256 columns
#define ROWS_PER_WG 256     // 8 waves * 32 rows (2 row-tiles of 16 per wave)

// workspace layout (float units)
#define WS_GMAX   0
#define WS_FACTOR (WS_GMAX + NQ_)
#define WS_ACC    (WS_FACTOR + B_N)
#define WS_PASS   (WS_ACC + 16)
#define PASS_STRIDE (B_N * NCHUNK * 2 + B_N)   // m[B*NC], s[B*NC], diag[B]

__global__ void init_ws_kernel(float* ws) {
    int i = blockIdx.x * blockDim.x + threadIdx.x;
    if (i < NQ_) ws[WS_GMAX + i] = 0.0f;
    if (i == 0)  ws[WS_ACC] = 0.0f;
}

// segment max of scores per question id (scores >= 0, int ordering == float ordering)
__global__ void seg_max_kernel(const float* __restrict__ scores,
                               const int* __restrict__ qid, float* ws) {
    int i = blockIdx.x * blockDim.x + threadIdx.x;
    if (i < B_N) {
        atomicMax((int*)(ws + WS_GMAX + qid[i]), __float_as_int(scores[i]));
    }
}

// factor[k] = (1 - 0.1*rank_k) * (gmax>0 ? 1 + 0.01*score_k/gmax : 1)
__global__ void factor_kernel(const float* __restrict__ scores,
                              const int* __restrict__ qid,
                              const int* __restrict__ ranks, float* ws) {
    int i = blockIdx.x * blockDim.x + threadIdx.x;
    if (i < B_N) {
        float g = ws[WS_GMAX + qid[i]];
        float f = 1.0f - RANK_W * (float)ranks[i];
        if (g > 0.0f) f *= 1.0f + SCORE_W * (scores[i] / g);
        ws[WS_FACTOR + i] = f;
    }
}

// Fused GEMM (X @ Y^T / TEMP) * weight, with online logsumexp over the column
// chunk per row, plus diagonal capture. Each wave computes a 32x64 register
// block (2 row-tiles x 4 col-tiles): 6 b64 loads per 8 WMMAs.
// pass==0: weight uses column factor (q->a direction)
// pass==1: weight uses row factor (a->q direction, i.e. transposed adj)
__global__ __launch_bounds__(256)
void gemm_lse_kernel(const float* __restrict__ X, const float* __restrict__ Y,
                     const int* __restrict__ qid, float* __restrict__ ws, int pass) {
    const float* __restrict__ factor = ws + WS_FACTOR;
    float* mOut = ws + WS_PASS + pass * PASS_STRIDE;
    float* sOut = mOut + B_N * NCHUNK;
    float* dOut = sOut + B_N * NCHUNK;

    const int wave  = threadIdx.x >> 5;
    const int lane  = threadIdx.x & 31;
    const int lh    = lane >> 4;       // which 16-lane half
    const int l16   = lane & 15;
    const int chunk = blockIdx.y;
    const int rbase = blockIdx.x * ROWS_PER_WG + wave * 32;  // 2 row-tiles of 16

    // per-row metadata: logical row index rr = rt*8 + r maps to
    // global row gi = rbase + rt*16 + r + 8*lh
    int   qidRow[16];
    float facRow[16];
#pragma unroll
    for (int rt = 0; rt < 2; ++rt)
#pragma unroll
        for (int r = 0; r < 8; ++r) {
            int gi = rbase + rt * 16 + r + 8 * lh;
            qidRow[rt * 8 + r] = qid[gi];
            facRow[rt * 8 + r] = factor[gi];
        }

    float m[16], s[16], dg[16];
#pragma unroll
    for (int rr = 0; rr < 16; ++rr) { m[rr] = -__builtin_inff(); s[rr] = 0.0f; dg[rr] = 0.0f; }

    // A fragment sources (f32 16x4 layout: lanes 0-15 hold K=0,1; 16-31 hold K=2,3)
    const float* xbase0 = X + (size_t)(rbase +      l16) * D_K + 2 * lh;
    const float* xbase1 = X + (size_t)(rbase + 16 + l16) * D_K + 2 * lh;

    for (int cs = 0; cs < 4; ++cs) {
        const int jbase = chunk * 256 + cs * 64;

        int   qidCol[4];
        float facCol[4];
#pragma unroll
        for (int t = 0; t < 4; ++t) {
            int gj = jbase + t * 16 + l16;
            qidCol[t] = qid[gj];
            facCol[t] = factor[gj];
        }

        v8f acc[2][4];
#pragma unroll
        for (int rt = 0; rt < 2; ++rt)
#pragma unroll
            for (int t = 0; t < 4; ++t)
                acc[rt][t] = (v8f){0.f, 0.f, 0.f, 0.f, 0.f, 0.f, 0.f, 0.f};

        const float* ybase = Y + (size_t)(jbase + l16) * D_K + 2 * lh;

#pragma unroll 4
        for (int k = 0; k < D_K; k += 4) {
            v2f a0 = *reinterpret_cast<const v2f*>(xbase0 + k);
            v2f a1 = *reinterpret_cast<const v2f*>(xbase1 + k);
#pragma unroll
            for (int t = 0; t < 4; ++t) {
                v2f b = *reinterpret_cast<const v2f*>(ybase + (size_t)t * 16 * D_K + k);
                acc[0][t] = __builtin_amdgcn_wmma_f32_16x16x4_f32(
                    false, a0, false, b, (short)0, acc[0][t], false, false);
                acc[1][t] = __builtin_amdgcn_wmma_f32_16x16x4_f32(
                    false, a1, false, b, (short)0, acc[1][t], false, false);
            }
        }

        // epilogue: weight, diag capture, online logsumexp fold (per row, 16 lanes)
#pragma unroll
        for (int rt = 0; rt < 2; ++rt)
#pragma unroll
            for (int t = 0; t < 4; ++t) {
                int gj = jbase + t * 16 + l16;
#pragma unroll
                for (int r = 0; r < 8; ++r) {
                    int rr = rt * 8 + r;
                    int gi = rbase + rt * 16 + r + 8 * lh;
                    float v = acc[rt][t][r] * INV_TEMP;
                    bool same = (qidRow[rr] == qidCol[t]) && (gi != gj);
                    float w = same ? (pass ? facRow[rr] : facCol[t]) : 1.0f;
                    v *= w;
                    if (gi == gj) dg[rr] = v;

                    float tm = v;
                    tm = fmaxf(tm, __shfl_xor(tm, 8, 16));
                    tm = fmaxf(tm, __shfl_xor(tm, 4, 16));
                    tm = fmaxf(tm, __shfl_xor(tm, 2, 16));
                    tm = fmaxf(tm, __shfl_xor(tm, 1, 16));
                    float nm = fmaxf(m[rr], tm);
                    float e = __expf(v - nm);
                    e += __shfl_xor(e, 8, 16);
                    e += __shfl_xor(e, 4, 16);
                    e += __shfl_xor(e, 2, 16);
                    e += __shfl_xor(e, 1, 16);
                    s[rr] = s[rr] * __expf(m[rr] - nm) + e;
                    m[rr] = nm;
                }
            }
    }

    // write per-row partials; diagonal is written by the chunk that owns column gi
#pragma unroll
    for (int rt = 0; rt < 2; ++rt)
#pragma unroll
        for (int r = 0; r < 8; ++r) {
            int rr = rt * 8 + r;
            float d = dg[rr];
            d += __shfl_xor(d, 8, 16);
            d += __shfl_xor(d, 4, 16);
            d += __shfl_xor(d, 2, 16);
            d += __shfl_xor(d, 1, 16);
            if (l16 == 0) {
                int gi = rbase + rt * 16 + r + 8 * lh;
                mOut[gi * NCHUNK + chunk] = m[rr];
                sOut[gi * NCHUNK + chunk] = s[rr];
                if ((gi >> 8) == chunk) dOut[gi] = d;
            }
        }
}

__global__ void combine_kernel(float* ws) {
    int i = blockIdx.x * blockDim.x + threadIdx.x;
    if (i >= 2 * B_N) return;
    int pass = i >> 12;          // B_N == 4096
    int row  = i & (B_N - 1);
    const float* mA = ws + WS_PASS + pass * PASS_STRIDE + row * NCHUNK;
    const float* sA = ws + WS_PASS + pass * PASS_STRIDE + B_N * NCHUNK + row * NCHUNK;
    float M = -__builtin_inff();
#pragma unroll
    for (int c = 0; c < NCHUNK; ++c) M = fmaxf(M, mA[c]);
    float S = 0.0f;
#pragma unroll
    for (int c = 0; c < NCHUNK; ++c) S += sA[c] * __expf(mA[c] - M);
    float d = ws[WS_PASS + pass * PASS_STRIDE + 2 * B_N * NCHUNK + row];
    float loss = M + __logf(S) - d;   // -log_softmax diagonal
    atomicAdd(ws + WS_ACC, loss);
}

__global__ void finalize_kernel(const float* ws, float* out) {
    out[0] = 0.5f * ws[WS_ACC] / (float)B_N;
}

extern "C" void kernel_launch(void* const* d_in, const int* in_sizes, int n_in,
                              void* d_out, int out_size, void* d_ws, size_t ws_size,
                              hipStream_t stream) {
    const float* q_emb  = (const float*)d_in[0];
    const float* a_emb  = (const float*)d_in[1];
    const int*   qid    = (const int*)d_in[2];
    const int*   ranks  = (const int*)d_in[3];
    const float* scores = (const float*)d_in[4];
    float* ws  = (float*)d_ws;
    float* out = (float*)d_out;

    init_ws_kernel<<<dim3(4), 256, 0, stream>>>(ws);
    seg_max_kernel<<<dim3(B_N / 256), 256, 0, stream>>>(scores, qid, ws);
    factor_kernel<<<dim3(B_N / 256), 256, 0, stream>>>(scores, qid, ranks, ws);

    dim3 grid(B_N / ROWS_PER_WG, NCHUNK);   // 16 x 16
    // pass 0: rows = q, cols = a, column-indexed weight  -> loss_q2a terms
    gemm_lse_kernel<<<grid, 256, 0, stream>>>(q_emb, a_emb, qid, ws, 0);
    // pass 1: rows = a, cols = q, row-indexed weight (adj^T) -> loss_a2q terms
    gemm_lse_kernel<<<grid, 256, 0, stream>>>(a_emb, q_emb, qid, ws, 1);

    combine_kernel<<<dim3(2 * B_N / 256), 256, 0, stream>>>(ws);
    finalize_kernel<<<1, 1, 0, stream>>>(ws, out);
}